// XDGRASP_47536698032139
// MI455X (gfx1250) — compile-verified
//
#include <hip/hip_runtime.h>
#include <math.h>

// ---------------------------------------------------------------------------
// XD-GRASP (toy) on MI455X / gfx1250.
// NUFFT basis A[k,h,w] = u[k,h]*v[k,w] (separable phase), u/v precomputed.
// Complex GEMMs on V_WMMA_F32_16X16X4_F32 (wave32) with re/im row-packing:
//   A-operand rows 0-7 = Re(data b=0..7), rows 8-15 = Im(data b=0..7)
//   P1 += Apack*B_re ; P2 += Apack*B_im        (2 WMMAs per K-chunk of 4)
//   out_re = P1.top - P2.bottom ; out_im = P2.top + P1.bottom
// top/bottom halves are 16 lanes apart in the C layout -> one shfl_xor(16)
// per accumulator VGPR in the epilogue. 100% matrix utilization.
// ---------------------------------------------------------------------------

#define T_   2
#define PH_  4
#define CH_  4
#define D_   2
#define SP_  4096
#define H_   64
#define W_   64
#define HW_  4096
#define P_   (T_ * PH_)    // 8 phases
#define B_   (CH_ * D_)    // 8 coil-images per phase

#define LAMBDA1 0.01f
#define LAMBDA2 0.01f
#define LR      0.5f
#define N_DC    ((float)(T_ * PH_ * CH_ * D_ * SP_))   // 262144
#define N_TVC   ((float)((T_ - 1) * PH_ * D_ * HW_))   // 32768
#define N_TVR   ((float)(T_ * (PH_ - 1) * D_ * HW_))   // 49152

typedef __attribute__((ext_vector_type(2))) float v2f;
typedef __attribute__((ext_vector_type(8))) float v8f;

__device__ __forceinline__ v8f wmma_f32(v2f a, v2f b, v8f c) {
  // D = A(16x4) * B(4x16) + C, fp32, wave32.
  return __builtin_amdgcn_wmma_f32_16x16x4_f32(false, a, false, b,
                                               (short)0, c, false, false);
}

__device__ __forceinline__ float2 unitc(float2 z) {
  float l = sqrtf(z.x * z.x + z.y * z.y);
  float s = (l > 0.0f) ? (1.0f / l) : 0.0f;
  return make_float2(z.x * s, z.y * s);
}

// ---------------------------------------------------------------------------
// 1) csm normalization over the D axis: csm_n = csm / sqrt(sum_d |csm|^2)
// ---------------------------------------------------------------------------
__global__ __launch_bounds__(256) void k_csm_norm(const float2* __restrict__ csm,
                                                  float2* __restrict__ csm_n) {
  int tid = blockIdx.x * blockDim.x + threadIdx.x;   // over T*CH*HW
  if (tid >= T_ * CH_ * HW_) return;
  int hw = tid % HW_;
  int c  = (tid / HW_) % CH_;
  int t  = tid / (HW_ * CH_);
  size_t i0 = (((size_t)(t * CH_ + c) * D_ + 0) * HW_) + hw;
  size_t i1 = i0 + HW_;
  float2 z0 = csm[i0], z1 = csm[i1];
  float s = z0.x * z0.x + z0.y * z0.y + z1.x * z1.x + z1.y * z1.y;
  float inv = 1.0f / sqrtf(s);
  csm_n[i0] = make_float2(z0.x * inv, z0.y * inv);
  csm_n[i1] = make_float2(z1.x * inv, z1.y * inv);
}

// ---------------------------------------------------------------------------
// 2) separable basis factors.
//    u_hk[p,h,k] = exp(-i*tr0[p,k]*(h-32))           (layout P,H,K)
//    v_wk[p,w,k] = exp(-i*tr1[p,k]*(w-32))/64        (layout P,W,K)
//    v_kw[p,k,w] = same values                        (layout P,K,W)
// ---------------------------------------------------------------------------
__global__ __launch_bounds__(256) void k_uv_gen(const float* __restrict__ traj,
                                                float2* __restrict__ u_hk,
                                                float2* __restrict__ v_wk,
                                                float2* __restrict__ v_kw) {
  int tid = blockIdx.x * blockDim.x + threadIdx.x;   // over P*64*SP
  if (tid >= P_ * 64 * SP_) return;
  int k = tid % SP_;
  int x = (tid / SP_) % 64;          // h or w
  int p = tid / (SP_ * 64);
  float tr0 = traj[(size_t)p * 2 * SP_ + k];
  float tr1 = traj[(size_t)p * 2 * SP_ + SP_ + k];
  float g = (float)(x - 32);
  float s0, c0, s1, c1;
  sincosf(tr0 * g, &s0, &c0);
  sincosf(tr1 * g, &s1, &c1);
  const float sc = 1.0f / 64.0f;     // 1/sqrt(H*W) folded into v
  u_hk[((size_t)p * H_ + x) * SP_ + k] = make_float2(c0, -s0);
  float2 v = make_float2(c1 * sc, -s1 * sc);
  v_wk[((size_t)p * W_ + x) * SP_ + k] = v;
  v_kw[((size_t)p * SP_ + k) * W_ + x] = v;
}

// ---------------------------------------------------------------------------
// 3) Adjoint NUFFT (fused basis): out[p,b,hw] = sum_k ksp[p,b,k]*conj(A[k,hw])
//    One wave per 16-wide hw tile (fixed h, 16 consecutive w).
//    Re/Im row-packed A operand; 2 WMMAs per K-chunk of 4.
// ---------------------------------------------------------------------------
__global__ __launch_bounds__(128) void k_adj_gemm(const float2* __restrict__ ksp,
                                                  const float2* __restrict__ u_hk,
                                                  const float2* __restrict__ v_wk,
                                                  float2* __restrict__ out) {
  int lane = threadIdx.x & 31;
  int tile = blockIdx.x * 4 + (threadIdx.x >> 5);   // 0 .. P_*256-1
  int p  = tile >> 8;
  int nt = tile & 255;
  int h     = nt >> 2;
  int wbase = (nt & 3) << 4;
  int m  = lane & 15;               // A row (b or b+8) / B col (n) of this lane
  int kk = (lane >> 4) << 1;        // K sub-offset within chunk: 0 or 2
  bool isRe = (m < 8);              // rows 0-7 carry Re, rows 8-15 carry Im

  const float4* arow = (const float4*)(ksp  + ((size_t)p * B_ + (m & 7)) * SP_);
  const float4* urow = (const float4*)(u_hk + ((size_t)p * H_ + h) * SP_);
  const float4* vrow = (const float4*)(v_wk + ((size_t)p * W_ + wbase + m) * SP_);

  v8f acc1 = {}, acc2 = {};         // P1 = Apack*Bre, P2 = Apack*Bim
#pragma unroll 4
  for (int k0 = 0; k0 < SP_; k0 += 4) {
    int q = (k0 + kk) >> 1;                 // float4 index (2 complex)
    float4 a  = arow[q];
    float4 uu = urow[q];
    float4 vv = vrow[q];
    v2f Apk = { isRe ? a.x : a.y, isRe ? a.z : a.w };
    // conj(u*v) for the two K values held by this lane
    float b0r =  uu.x * vv.x - uu.y * vv.y;
    float b0i = -(uu.x * vv.y + uu.y * vv.x);
    float b1r =  uu.z * vv.z - uu.w * vv.w;
    float b1i = -(uu.z * vv.w + uu.w * vv.z);
    v2f Bre = { b0r, b1r };
    v2f Bim = { b0i, b1i };
    acc1 = wmma_f32(Apk, Bre, acc1);
    acc2 = wmma_f32(Apk, Bim, acc2);
  }
  // recombine top (Ar*X, rows 0-7 -> lanes 0-15) with bottom (Ai*X, rows 8-15
  // -> lanes 16-31): full-EXEC shuffles, then lanes 0-15 write b=0..7.
  float re[8], im[8];
#pragma unroll
  for (int j = 0; j < 8; ++j) {
    float p1x = __shfl_xor((float)acc1[j], 16, 32);   // Ai*Br for lanes < 16
    float p2x = __shfl_xor((float)acc2[j], 16, 32);   // Ai*Bi for lanes < 16
    re[j] = acc1[j] - p2x;
    im[j] = acc2[j] + p1x;
  }
  if (lane < 16) {
    size_t base = (size_t)p * B_ * HW_ + (size_t)h * W_ + wbase + lane;
#pragma unroll
    for (int j = 0; j < 8; ++j)
      out[base + (size_t)j * HW_] = make_float2(re[j], im[j]);
  }
}

// ---------------------------------------------------------------------------
// 4) Forward NUFFT (fused basis) + data-consistency residual:
//    dc[p,b,ksp] = sum_hw img[p,b,hw]*A[ksp,hw] - ks[p,b,ksp]
// ---------------------------------------------------------------------------
__global__ __launch_bounds__(128) void k_fwd_gemm(const float2* __restrict__ img,
                                                  const float2* __restrict__ u_hk,
                                                  const float2* __restrict__ v_kw,
                                                  const float2* __restrict__ ks,
                                                  float2* __restrict__ dc) {
  int lane = threadIdx.x & 31;
  int tile = blockIdx.x * 4 + (threadIdx.x >> 5);
  int p  = tile >> 8;
  int nt = tile & 255;
  int kbase = nt << 4;
  int m  = lane & 15;
  int kk = (lane >> 4) << 1;
  bool isRe = (m < 8);
  int kcol = kbase + m;             // spoke index: this lane's B column

  const float4* arow = (const float4*)(img + ((size_t)p * B_ + (m & 7)) * HW_);
  const float2* ucol = u_hk + (size_t)p * H_ * SP_ + kcol;                // [h*SP_]
  const float4* vrow = (const float4*)(v_kw + ((size_t)p * SP_ + kcol) * W_);

  v8f acc1 = {}, acc2 = {};
  for (int h = 0; h < H_; ++h) {
    float2 uc = ucol[(size_t)h * SP_];
#pragma unroll 4
    for (int wc = 0; wc < W_; wc += 4) {
      int hw0 = h * W_ + wc;
      float4 a  = arow[(hw0 + kk) >> 1];
      float4 vv = vrow[(wc + kk) >> 1];
      v2f Apk = { isRe ? a.x : a.y, isRe ? a.z : a.w };
      // B[hw, kcol] = u[kcol,h]*v[kcol,w]  (no conj on forward)
      float b0r = uc.x * vv.x - uc.y * vv.y;
      float b0i = uc.x * vv.y + uc.y * vv.x;
      float b1r = uc.x * vv.z - uc.y * vv.w;
      float b1i = uc.x * vv.w + uc.y * vv.z;
      v2f Bre = { b0r, b1r };
      v2f Bim = { b0i, b1i };
      acc1 = wmma_f32(Apk, Bre, acc1);
      acc2 = wmma_f32(Apk, Bim, acc2);
    }
  }
  float re[8], im[8];
#pragma unroll
  for (int j = 0; j < 8; ++j) {
    float p1x = __shfl_xor((float)acc1[j], 16, 32);
    float p2x = __shfl_xor((float)acc2[j], 16, 32);
    re[j] = acc1[j] - p2x;
    im[j] = acc2[j] + p1x;
  }
  if (lane < 16) {
    size_t base = (size_t)p * B_ * SP_ + kbase + lane;
#pragma unroll
    for (int j = 0; j < 8; ++j) {
      float2 kd = ks[base + (size_t)j * SP_];
      dc[base + (size_t)j * SP_] = make_float2(re[j] - kd.x, im[j] - kd.y);
    }
  }
}

// ---------------------------------------------------------------------------
// 5) coil-combine with conj(csm_n): image_init = (1/5) sum_c conj(csm)*adj
// ---------------------------------------------------------------------------
__global__ __launch_bounds__(256) void k_combine_init(const float2* __restrict__ adj,
                                                      const float2* __restrict__ csm_n,
                                                      float2* __restrict__ params,
                                                      float2* __restrict__ out_init) {
  int tid = blockIdx.x * blockDim.x + threadIdx.x;   // over T*PH*D*HW
  if (tid >= P_ * D_ * HW_) return;
  int hw = tid % HW_;
  int d  = (tid / HW_) % D_;
  int p  = tid / (HW_ * D_);
  int t  = p / PH_;
  float re = 0.0f, im = 0.0f;
#pragma unroll
  for (int c = 0; c < CH_; ++c) {
    float2 s = csm_n[(((size_t)(t * CH_ + c) * D_ + d) * HW_) + hw];
    float2 a = adj[((size_t)p * B_ + c * D_ + d) * HW_ + hw];
    re += s.x * a.x + s.y * a.y;     // conj(s)*a
    im += s.x * a.y - s.y * a.x;
  }
  float2 v = make_float2(re * 0.2f, im * 0.2f);
  params[tid]   = v;
  out_init[tid] = v;
}

// ---------------------------------------------------------------------------
// 6) coil-expand: img_mc[p, c*D+d, hw] = params[p,d,hw] * csm_n[t,c,d,hw]
// ---------------------------------------------------------------------------
__global__ __launch_bounds__(256) void k_expand(const float2* __restrict__ params,
                                                const float2* __restrict__ csm_n,
                                                float2* __restrict__ img_mc) {
  int tid = blockIdx.x * blockDim.x + threadIdx.x;   // over P*B*HW
  if (tid >= P_ * B_ * HW_) return;
  int hw = tid % HW_;
  int b  = (tid / HW_) % B_;
  int p  = tid / (HW_ * B_);
  int d  = b % D_;
  int c  = b / D_;
  int t  = p / PH_;
  float2 pv = params[((size_t)p * D_ + d) * HW_ + hw];
  float2 s  = csm_n[(((size_t)(t * CH_ + c) * D_ + d) * HW_) + hw];
  img_mc[tid] = make_float2(pv.x * s.x - pv.y * s.y, pv.x * s.y + pv.y * s.x);
}

// ---------------------------------------------------------------------------
// 7) gradient step: params_out = params_in - LR*(E^H(dc)/N_dc + TV grads)
// ---------------------------------------------------------------------------
__global__ __launch_bounds__(256) void k_grad_step(const float2* __restrict__ gadj,
                                                   const float2* __restrict__ csm_n,
                                                   const float2* __restrict__ pin,
                                                   float2* __restrict__ pout) {
  int tid = blockIdx.x * blockDim.x + threadIdx.x;   // over T*PH*D*HW
  if (tid >= P_ * D_ * HW_) return;
  int hw = tid % HW_;
  int d  = (tid / HW_) % D_;
  int p  = tid / (HW_ * D_);
  int t  = p / PH_;
  int ph = p % PH_;

  // data-consistency gradient: sum_c conj(csm)*adj(dc) / N_dc
  float gr = 0.0f, gi = 0.0f;
#pragma unroll
  for (int c = 0; c < CH_; ++c) {
    float2 s = csm_n[(((size_t)(t * CH_ + c) * D_ + d) * HW_) + hw];
    float2 a = gadj[((size_t)p * B_ + c * D_ + d) * HW_ + hw];
    gr += s.x * a.x + s.y * a.y;
    gi += s.x * a.y - s.y * a.x;
  }
  gr *= (1.0f / N_DC);
  gi *= (1.0f / N_DC);

  const int strT  = PH_ * D_ * HW_;
  const int strPH = D_ * HW_;
  float2 cur = pin[tid];

  // contrast TV (along t)
  if (t + 1 < T_) {
    float2 nb = pin[tid + strT];
    float2 u = unitc(make_float2(nb.x - cur.x, nb.y - cur.y));
    gr -= (LAMBDA1 / N_TVC) * u.x;  gi -= (LAMBDA1 / N_TVC) * u.y;
  }
  if (t >= 1) {
    float2 nb = pin[tid - strT];
    float2 u = unitc(make_float2(cur.x - nb.x, cur.y - nb.y));
    gr += (LAMBDA1 / N_TVC) * u.x;  gi += (LAMBDA1 / N_TVC) * u.y;
  }
  // respiratory TV (along ph)
  if (ph + 1 < PH_) {
    float2 nb = pin[tid + strPH];
    float2 u = unitc(make_float2(nb.x - cur.x, nb.y - cur.y));
    gr -= (LAMBDA2 / N_TVR) * u.x;  gi -= (LAMBDA2 / N_TVR) * u.y;
  }
  if (ph >= 1) {
    float2 nb = pin[tid - strPH];
    float2 u = unitc(make_float2(cur.x - nb.x, cur.y - nb.y));
    gr += (LAMBDA2 / N_TVR) * u.x;  gi += (LAMBDA2 / N_TVR) * u.y;
  }
  pout[tid] = make_float2(cur.x - LR * gr, cur.y - LR * gi);
}

// ---------------------------------------------------------------------------
// host-side orchestration
// ---------------------------------------------------------------------------
extern "C" void kernel_launch(void* const* d_in, const int* in_sizes, int n_in,
                              void* d_out, int out_size, void* d_ws, size_t ws_size,
                              hipStream_t stream) {
  const float2* ks   = (const float2*)d_in[0];   // kspace_data
  const float2* ksc  = (const float2*)d_in[1];   // kspace_data_compensated
  const float*  traj = (const float*)d_in[2];    // kspace_traj
  const float2* csm  = (const float2*)d_in[3];   // csm
  float2* out = (float2*)d_out;                  // [params | image_init], 131072 cplx

  // workspace carve (256B aligned)
  char* w = (char*)d_ws;
  auto carve = [&](size_t n_cplx) {
    void* r = (void*)w;
    w += (n_cplx * sizeof(float2) + 255) & ~(size_t)255;
    return (float2*)r;
  };
  float2* csm_n  = carve((size_t)T_ * CH_ * D_ * HW_);   // 0.5 MB
  float2* u_hk   = carve((size_t)P_ * H_ * SP_);         // 16 MB
  float2* v_wk   = carve((size_t)P_ * W_ * SP_);         // 16 MB
  float2* v_kw   = carve((size_t)P_ * SP_ * W_);         // 16 MB
  float2* adjbuf = carve((size_t)P_ * B_ * HW_);         // 2 MB
  float2* img_mc = carve((size_t)P_ * B_ * HW_);         // 2 MB
  float2* dcbuf  = carve((size_t)P_ * B_ * SP_);         // 2 MB
  float2* pA     = carve((size_t)P_ * D_ * HW_);         // 0.5 MB
  float2* pB     = carve((size_t)P_ * D_ * HW_);         // 0.5 MB

  const int EW = 256;
  // 1) csm normalize
  k_csm_norm<<<(T_ * CH_ * HW_ + EW - 1) / EW, EW, 0, stream>>>(csm, csm_n);
  // 2) basis factors
  k_uv_gen<<<(P_ * 64 * SP_ + EW - 1) / EW, EW, 0, stream>>>(traj, u_hk, v_wk, v_kw);

  const int GEMM_BLOCKS = (P_ * 256) / 4;   // 4 waves/block, 1 tile/wave
  // 3) image_init = combine(adjoint(ksc)) / 5   -> pA and out[65536..]
  k_adj_gemm<<<GEMM_BLOCKS, 128, 0, stream>>>(ksc, u_hk, v_wk, adjbuf);
  k_combine_init<<<(P_ * D_ * HW_ + EW - 1) / EW, EW, 0, stream>>>(
      adjbuf, csm_n, pA, out + (size_t)P_ * D_ * HW_);

  // 4) two gradient-descent iterations (ping-pong pA <-> pB)
  float2* cur = pA;
  float2* nxt = pB;
  for (int it = 0; it < 2; ++it) {
    k_expand<<<(P_ * B_ * HW_ + EW - 1) / EW, EW, 0, stream>>>(cur, csm_n, img_mc);
    k_fwd_gemm<<<GEMM_BLOCKS, 128, 0, stream>>>(img_mc, u_hk, v_kw, ks, dcbuf);
    k_adj_gemm<<<GEMM_BLOCKS, 128, 0, stream>>>(dcbuf, u_hk, v_wk, adjbuf);
    k_grad_step<<<(P_ * D_ * HW_ + EW - 1) / EW, EW, 0, stream>>>(adjbuf, csm_n, cur, nxt);
    float2* tmp = cur; cur = nxt; nxt = tmp;
  }
  // 5) final params -> out[0..]
  hipMemcpyAsync(out, cur, (size_t)P_ * D_ * HW_ * sizeof(float2),
                 hipMemcpyDeviceToDevice, stream);
}